// NoiseRegressionEval_28303834481267
// MI455X (gfx1250) — compile-verified
//
#include <hip/hip_runtime.h>
#include <hip/hip_bf16.h>
#include <stdint.h>

// ---------------- problem constants ----------------
#define N0    2048
#define NTOT  16384          // 2^3 replicates * N0
#define KNN   12
#define NTILES (NTOT / 16)   // 1024 column tiles of 16

typedef float v2f __attribute__((ext_vector_type(2)));
typedef float v8f __attribute__((ext_vector_type(8)));

// ---------------- deterministic PRNG (threefry2x32, key = (0,42)) ----------------
__device__ __forceinline__ uint32_t rotl32(uint32_t x, int r) {
  return (x << r) | (x >> (32 - r));
}

__device__ __forceinline__ void threefry2x32(uint32_t c0, uint32_t c1,
                                             uint32_t& o0, uint32_t& o1) {
  const uint32_t k0 = 0u, k1 = 42u;
  const uint32_t k2 = k0 ^ k1 ^ 0x1BD11BDAu;
  uint32_t x0 = c0 + k0, x1 = c1 + k1;
  const int ra[4] = {13, 15, 26, 6};
  const int rb[4] = {17, 29, 16, 24};
#pragma unroll
  for (int i = 0; i < 4; ++i) { x0 += x1; x1 = rotl32(x1, ra[i]); x1 ^= x0; }
  x0 += k1; x1 += k2 + 1u;
#pragma unroll
  for (int i = 0; i < 4; ++i) { x0 += x1; x1 = rotl32(x1, rb[i]); x1 ^= x0; }
  x0 += k2; x1 += k0 + 2u;
#pragma unroll
  for (int i = 0; i < 4; ++i) { x0 += x1; x1 = rotl32(x1, ra[i]); x1 ^= x0; }
  x0 += k0; x1 += k1 + 3u;
#pragma unroll
  for (int i = 0; i < 4; ++i) { x0 += x1; x1 = rotl32(x1, rb[i]); x1 ^= x0; }
  x0 += k1; x1 += k2 + 4u;
#pragma unroll
  for (int i = 0; i < 4; ++i) { x0 += x1; x1 = rotl32(x1, ra[i]); x1 ^= x0; }
  x0 += k2; x1 += k0 + 5u;
  o0 = x0; o1 = x1;
}

__device__ __forceinline__ float gauss_noise(uint32_t i, uint32_t j) {
  uint32_t r0, r1;
  threefry2x32(i, j, r0, r1);
  float u1 = (float)(r0 >> 8) * (1.0f / 16777216.0f) + (1.0f / 33554432.0f);
  float u2 = (float)(r1 >> 8) * (1.0f / 16777216.0f);
  return sqrtf(-2.0f * logf(u1)) * __cosf(6.28318530718f * u2);
}

// ---------------- kernel 1: prep (frac -> supercell -> noise -> cart) ----------------
// Stores positions as (x, y, z, -0.5*|p|^2) so the column norm can ride in the
// K=4 slot of the WMMA and d^2 = fma(-2, S, |x|^2) needs no extra loads.
__global__ __launch_bounds__(256)
void prep_kernel(const float* __restrict__ pos, const float* __restrict__ cell,
                 const int* __restrict__ numbers,
                 float* __restrict__ cart4,
                 float* __restrict__ accum, float* __restrict__ out_nums) {
  const int i = blockIdx.x * 256 + threadIdx.x;

  // cell is 3x3 row-major; compute inverse (uniform per thread, cheap)
  float m[9];
#pragma unroll
  for (int t = 0; t < 9; ++t) m[t] = cell[t];
  const float c00 = m[4] * m[8] - m[5] * m[7];
  const float c01 = m[5] * m[6] - m[3] * m[8];
  const float c02 = m[3] * m[7] - m[4] * m[6];
  const float idet = 1.0f / (m[0] * c00 + m[1] * c01 + m[2] * c02);
  float inv[9];
  inv[0] = c00 * idet;
  inv[1] = (m[2] * m[7] - m[1] * m[8]) * idet;
  inv[2] = (m[1] * m[5] - m[2] * m[4]) * idet;
  inv[3] = c01 * idet;
  inv[4] = (m[0] * m[8] - m[2] * m[6]) * idet;
  inv[5] = (m[2] * m[3] - m[0] * m[5]) * idet;
  inv[6] = c02 * idet;
  inv[7] = (m[1] * m[6] - m[0] * m[7]) * idet;
  inv[8] = (m[0] * m[4] - m[1] * m[3]) * idet;

  float eps2 = 0.0f;
  if (i < NTOT) {
    const int base = i & (N0 - 1);
    const int s = i >> 11;  // supercell shift index, row-major (ij) meshgrid
    const float sx = (float)((s >> 2) & 1);
    const float sy = (float)((s >> 1) & 1);
    const float sz = (float)(s & 1);

    const float px = pos[base * 3 + 0];
    const float py = pos[base * 3 + 1];
    const float pz = pos[base * 3 + 2];

    float f0 = px * inv[0] + py * inv[3] + pz * inv[6] + sx;
    float f1 = px * inv[1] + py * inv[4] + pz * inv[7] + sy;
    float f2 = px * inv[2] + py * inv[5] + pz * inv[8] + sz;

    const float e0 = 0.05f * gauss_noise((uint32_t)i, 0u);
    const float e1 = 0.05f * gauss_noise((uint32_t)i, 1u);
    const float e2 = 0.05f * gauss_noise((uint32_t)i, 2u);
    eps2 = e0 * e0 + e1 * e1 + e2 * e2;
    f0 += e0; f1 += e1; f2 += e2;

    const float cx = f0 * m[0] + f1 * m[3] + f2 * m[6];
    const float cy = f0 * m[1] + f1 * m[4] + f2 * m[7];
    const float cz = f0 * m[2] + f1 * m[5] + f2 * m[8];

    cart4[4 * i + 0] = cx;
    cart4[4 * i + 1] = cy;
    cart4[4 * i + 2] = cz;
    cart4[4 * i + 3] = -0.5f * (cx * cx + cy * cy + cz * cz);
    out_nums[i] = (float)numbers[base];
  }

  // block reduction of eps^2 -> global accumulator
  __shared__ float red[256];
  red[threadIdx.x] = eps2;
  __syncthreads();
#pragma unroll
  for (int off = 128; off > 0; off >>= 1) {
    if (threadIdx.x < off) red[threadIdx.x] += red[threadIdx.x + off];
    __syncthreads();
  }
  if (threadIdx.x == 0) atomicAdd(accum, red[0]);
}

// ---------------- kernel 2: WMMA pairwise d^2 + per-row top-12 ----------------
// One workgroup (8 wave32s) handles 16 output rows. Per column tile:
//   A (16x4) = column block y with K = (y.x, y.y, y.z, -0.5|y|^2)
//   B (4x16) = row block x with K = (x.x, x.y, x.z, 1)
//   D[M=col][N=row] = x.y - 0.5|y|^2  -> d^2 = |x|^2 - 2*D
// Each lane's 8 acc VGPRs all belong to ONE row (N = lane&15); lanes n and
// n+16 cover that row's two column halves (cols cb + r + 8*(lane>>4)).
// The column-tile load is software-pipelined one iteration ahead so the
// s_wait_loadcnt lands behind the selection work of the previous tile.
__global__ __launch_bounds__(256)
void knn_kernel(const float4* __restrict__ cart4,
                float* __restrict__ out_dist, float* __restrict__ out_idx) {
  const int tid  = threadIdx.x;
  const int wave = tid >> 5;
  const int lane = tid & 31;
  const int n    = lane & 15;   // row within block (N index of D)
  const int hi   = lane >> 4;   // K-half for A/B operands, col-half of D
  const int rb   = blockIdx.x * 16;
  const int rowg = rb + n;

  // B operand (fixed for the whole sweep): row positions x, K3 = 1
  const float4 xr = cart4[rowg];
  v2f b;
  b.x = hi ? xr.z : xr.x;
  b.y = hi ? 1.0f : xr.y;
  const float sqx = -2.0f * xr.w;   // |x|^2

  float best_d[KNN];
  int   best_i[KNN];
#pragma unroll
  for (int k = 0; k < KNN; ++k) { best_d[k] = __builtin_inff(); best_i[k] = -1; }

  // software-pipelined column sweep
  float4 yc = cart4[wave * 16 + n];
  for (int t = wave; t < NTILES; t += 8) {
    const int cb = t * 16;
    // prefetch next tile (clamped so the load is unconditional; EXEC stays full)
    const int tn = t + 8;
    const float4 ycn = cart4[((tn < NTILES) ? tn * 16 : 0) + n];

    v2f a;
    a.x = hi ? yc.z : yc.x;
    a.y = hi ? yc.w : yc.y;

    v8f acc = {};
    acc = __builtin_amdgcn_wmma_f32_16x16x4_f32(
        /*neg_a=*/false, a, /*neg_b=*/false, b,
        /*c_mod=*/(short)0, acc, /*reuse_a=*/false, /*reuse_b=*/false);

#pragma unroll
    for (int r = 0; r < 8; ++r) {
      const int colg = cb + r + 8 * hi;
      float d2v = __builtin_fmaf(-2.0f, acc[r], sqx);
      d2v = (colg == rowg) ? __builtin_inff() : d2v;   // exclude self
      if (d2v < best_d[KNN - 1]) {
        // select-based sorted insert (ascending); pure min/max/cndmask chain
        float cd = d2v; int ci = colg;
#pragma unroll
        for (int k = 0; k < KNN; ++k) {
          const bool sw = cd < best_d[k];
          const float lo = fminf(cd, best_d[k]);
          const float hv = fmaxf(cd, best_d[k]);
          const int li = sw ? ci : best_i[k];
          const int hk = sw ? best_i[k] : ci;
          best_d[k] = lo; best_i[k] = li;
          cd = hv; ci = hk;
        }
      }
    }
    yc = ycn;
  }

  // merge 16 partial lists per row through LDS
  __shared__ float s_d[16][192];
  __shared__ int   s_i[16][192];
  const int list = (wave * 2 + hi) * KNN;
#pragma unroll
  for (int k = 0; k < KNN; ++k) {
    s_d[n][list + k] = best_d[k];
    s_i[n][list + k] = best_i[k];
  }
  __syncthreads();

  if (tid < 16) {
    const int row = tid;
    const int rg = rb + row;
    for (int k = 0; k < KNN; ++k) {
      float bm = __builtin_inff();
      int be = 0;
      for (int e = 0; e < 192; ++e) {
        const float v = s_d[row][e];
        if (v < bm) { bm = v; be = e; }
      }
      const int gi = s_i[row][be];
      s_d[row][be] = __builtin_inff();
      out_dist[rg * KNN + k] = sqrtf(fmaxf(bm, 0.0f));
      out_idx[rg * KNN + k]  = (float)gi;
    }
  }
}

// ---------------- kernel 3: finalize noise scalar ----------------
__global__ void fin_kernel(const float* __restrict__ accum, float* __restrict__ out_noise) {
  out_noise[0] = sqrtf(accum[0] / (3.0f * (float)NTOT));
}

// ---------------- launcher ----------------
extern "C" void kernel_launch(void* const* d_in, const int* in_sizes, int n_in,
                              void* d_out, int out_size, void* d_ws, size_t ws_size,
                              hipStream_t stream) {
  const float* pos     = (const float*)d_in[0];  // [2048,3]
  const float* cell    = (const float*)d_in[1];  // [3,3]
  const int*   numbers = (const int*)d_in[2];    // [2048]
  float* out = (float*)d_out;
  // output layout: knn_dist [NTOT*12] | knn_idx [NTOT*12] | numbers_tiled [NTOT] | noise [1]
  float* out_dist = out;
  float* out_idx  = out + NTOT * KNN;
  float* out_nums = out + 2 * NTOT * KNN;
  float* out_nz   = out + 2 * NTOT * KNN + NTOT;

  float* cart4 = (float*)d_ws;             // NTOT * 4 floats (x, y, z, -0.5|p|^2)
  float* accum = cart4 + 4 * NTOT;         // 1 float

  hipMemsetAsync(accum, 0, sizeof(float), stream);
  prep_kernel<<<NTOT / 256, 256, 0, stream>>>(pos, cell, numbers, cart4, accum, out_nums);
  knn_kernel<<<NTOT / 16, 256, 0, stream>>>((const float4*)cart4, out_dist, out_idx);
  fin_kernel<<<1, 1, 0, stream>>>(accum, out_nz);
}